// FGBackdoor_19911468384622
// MI455X (gfx1250) — compile-verified
//
#include <hip/hip_runtime.h>
#include <stdint.h>

#define CHANNELS 64
#define MAXD 128

// ---------- CDNA5 async global->LDS helpers (gfx1250) ----------
__device__ __forceinline__ void async_copy_b128(uint32_t lds_byte_addr, uint64_t gaddr) {
    // GLOBAL_LOAD_ASYNC_TO_LDS_B128, GV mode: vdst = LDS byte address, vaddr = 64-bit global addr
    asm volatile("global_load_async_to_lds_b128 %0, %1, off"
                 :: "v"(lds_byte_addr), "v"(gaddr)
                 : "memory");
}

__device__ __forceinline__ void wait_async_zero() {
    asm volatile("s_wait_asynccnt 0x0" ::: "memory");
}

// ---------- kernel 1: zero ints ----------
__global__ void zero_i32_kernel(int* __restrict__ a, int n) {
    int i = blockIdx.x * blockDim.x + threadIdx.x;
    if (i < n) a[i] = 0;
}

// ---------- kernel 2: destination histogram ----------
__global__ void hist_kernel(const int* __restrict__ col, int* __restrict__ deg, int e) {
    int i = blockIdx.x * blockDim.x + threadIdx.x;
    if (i < e) atomicAdd(&deg[col[i]], 1);
}

// ---------- kernel 3: exclusive scan (single block, Hillis-Steele over chunks) ----------
__global__ __launch_bounds__(1024) void scan_kernel(const int* __restrict__ deg,
                                                    int* __restrict__ starts, int n) {
    __shared__ int tmp[1024];
    __shared__ int carry;
    if (threadIdx.x == 0) carry = 0;
    __syncthreads();
    for (int base = 0; base < n; base += 1024) {
        int i = base + (int)threadIdx.x;
        int v = (i < n) ? deg[i] : 0;
        tmp[threadIdx.x] = v;
        __syncthreads();
        for (int off = 1; off < 1024; off <<= 1) {
            int u = (threadIdx.x >= (unsigned)off) ? tmp[threadIdx.x - off] : 0;
            __syncthreads();
            tmp[threadIdx.x] += u;
            __syncthreads();
        }
        int incl = tmp[threadIdx.x];
        if (i < n) starts[i] = carry + incl - v;   // exclusive
        __syncthreads();
        if (threadIdx.x == 1023) carry += tmp[1023];
        __syncthreads();
    }
}

// ---------- kernel 4: counting-sort scatter into CSR order ----------
__global__ void scatter_kernel(const int* __restrict__ row, const int* __restrict__ col,
                               const float* __restrict__ w, const int* __restrict__ starts,
                               int* __restrict__ cursor, int* __restrict__ srow,
                               float* __restrict__ sw, int e) {
    int i = blockIdx.x * blockDim.x + threadIdx.x;
    if (i < e) {
        int c = col[i];
        int p = atomicAdd(&cursor[c], 1);
        int idx = starts[c] + p;
        srow[idx] = row[i];
        sw[idx]   = w[i];
    }
}

// ---------- kernel 5: per-node per-channel lower median ----------
// One block (64 threads = 2 waves, wave32) per node; lane c owns channel c.
__global__ __launch_bounds__(CHANNELS) void median_kernel(
    const float* __restrict__ x, const int* __restrict__ starts,
    const int* __restrict__ deg, const int* __restrict__ srow,
    const float* __restrict__ sw, float* __restrict__ out, int n) {

    // 16B-aligned: async B128 writes require natively aligned LDS destinations
    __shared__ __align__(16) float vals[MAXD * CHANNELS];  // 32 KB staged neighbor rows
    __shared__ int   rowS[MAXD];
    __shared__ float wS[MAXD];

    int node = blockIdx.x;
    if (node >= n) return;
    int c = (int)threadIdx.x;                 // channel 0..63 (d is uniform per block)
    int d  = deg[node];
    int s0 = starts[node];

    if (d == 0) { out[(size_t)node * CHANNELS + c] = 0.0f; return; }

    if (d <= MAXD) {
        // stage edge metadata
        for (int k = c; k < d; k += CHANNELS) { rowS[k] = srow[s0 + k]; wS[k] = sw[s0 + k]; }
        __syncthreads();

        // async DMA neighbor rows of x into LDS: d rows * 256B, b128 chunks (16B aligned both sides)
        int chunks = d * (CHANNELS / 4);      // 16 x b128 per 64-float row
        for (int q = c; q < chunks; q += CHANNELS) {
            int k = q >> 4;                   // neighbor row
            int e4 = q & 15;                  // 4-float group within the row
            uint64_t g = (uint64_t)(uintptr_t)(x + (size_t)rowS[k] * CHANNELS + e4 * 4);
            uint32_t la = (uint32_t)(uintptr_t)(&vals[q * 4]);
            async_copy_b128(la, g);
        }
        wait_async_zero();                    // this wave's async ops complete
        __syncthreads();                      // all waves' staging complete

        // fold edge weights in (lane c owns column c exclusively -> no barrier needed after)
        for (int k = 0; k < d; ++k) vals[k * CHANNELS + c] *= wS[k];

        // counting selection: rank t = (d-1)/2, stable tie-break by index.
        // Lane c reads only column c -> always bank c -> conflict-free.
        int t = (d - 1) >> 1;
        float med = 0.0f;
        for (int i = 0; i < d; ++i) {
            float vi = vals[i * CHANNELS + c];
            int cnt = 0;
            for (int j = 0; j < d; ++j) {
                float vj = vals[j * CHANNELS + c];
                cnt += (vj < vi) || (vj == vi && j < i);
            }
            if (cnt == t) med = vi;
        }
        out[(size_t)node * CHANNELS + c] = med;
    } else {
        // fallback (statistically dead for avg degree 16, kept for correctness):
        // selection straight from global memory
        int t = (d - 1) >> 1;
        float med = 0.0f;
        for (int i = 0; i < d; ++i) {
            float vi = x[(size_t)srow[s0 + i] * CHANNELS + c] * sw[s0 + i];
            int cnt = 0;
            for (int j = 0; j < d; ++j) {
                float vj = x[(size_t)srow[s0 + j] * CHANNELS + c] * sw[s0 + j];
                cnt += (vj < vi) || (vj == vi && j < i);
            }
            if (cnt == t) med = vi;
        }
        out[(size_t)node * CHANNELS + c] = med;
    }
}

extern "C" void kernel_launch(void* const* d_in, const int* in_sizes, int n_in,
                              void* d_out, int out_size, void* d_ws, size_t ws_size,
                              hipStream_t stream) {
    const float* x  = (const float*)d_in[0];   // [N, 64]
    const int*   ei = (const int*)d_in[1];     // [2, E]
    const float* ew = (const float*)d_in[2];   // [E]
    float* out = (float*)d_out;

    const int E = in_sizes[2];
    const int N = in_sizes[0] / CHANNELS;

    const int* row = ei;       // edge_index[0,:]
    const int* col = ei + E;   // edge_index[1,:]

    // workspace layout: deg[N] | cursor[N] | starts[N] | srow[E] | sw[E]  (~7 MB)
    int*   deg    = (int*)d_ws;
    int*   cursor = deg + N;
    int*   starts = cursor + N;
    int*   srow   = starts + N;
    float* sw     = (float*)(srow + E);

    zero_i32_kernel<<<(2 * N + 255) / 256, 256, 0, stream>>>(deg, 2 * N); // deg + cursor
    hist_kernel<<<(E + 255) / 256, 256, 0, stream>>>(col, deg, E);
    scan_kernel<<<1, 1024, 0, stream>>>(deg, starts, N);
    scatter_kernel<<<(E + 255) / 256, 256, 0, stream>>>(row, col, ew, starts, cursor,
                                                        srow, sw, E);
    median_kernel<<<N, CHANNELS, 0, stream>>>(x, starts, deg, srow, sw, out, N);
}